// DynamicViewSampler_58385785422189
// MI455X (gfx1250) — compile-verified
//
#include <hip/hip_runtime.h>
#include <hip/hip_bf16.h>

#define NUM_VIEWS 8
#define GAMMA 10.0f
#define EPSV 1e-6f
#define B_   16
#define LMAX 4096
#define D_   1024

#define CHUNK 64
#define COLS  64            // columns per block
#define SV_STRIDE 72        // 2*72 % 64 == 16 -> half-wave bank windows disjoint (B frags)
#define SM_STRIDE 68        // 4r vs 4r+2 dword banks -> half-waves disjoint (A frags)
#define NCB (D_/COLS)       // 16 column blocks per batch
#define ASYNC_PER_THREAD 8  // CHUNK*COLS*4B / 16B / 128 threads

typedef float v2f __attribute__((ext_vector_type(2)));
typedef float v8f __attribute__((ext_vector_type(8)));

#if defined(__HIP_DEVICE_COMPILE__) && defined(__AMDGCN__)
#if !__has_builtin(__builtin_amdgcn_wmma_f32_16x16x4_f32)
#error "device pass: missing __builtin_amdgcn_wmma_f32_16x16x4_f32"
#endif
#endif

// ---- CDNA5 async global->LDS copy (ASYNCcnt-tracked), 16B per lane ----
__device__ __forceinline__ void async_load_b128(unsigned lds_off, const float* gptr) {
    asm volatile("global_load_async_to_lds_b128 %0, %1, off"
                 :: "v"(lds_off), "v"((unsigned long long)(size_t)gptr)
                 : "memory");
}
__device__ __forceinline__ void wait_asynccnt_8() {
#if __has_builtin(__builtin_amdgcn_s_wait_asynccnt)
    __builtin_amdgcn_s_wait_asynccnt(8);
#else
    asm volatile("s_wait_asynccnt 0x8" ::: "memory");
#endif
}
__device__ __forceinline__ void wait_asynccnt_0() {
#if __has_builtin(__builtin_amdgcn_s_wait_asynccnt)
    __builtin_amdgcn_s_wait_asynccnt(0);
#else
    asm volatile("s_wait_asynccnt 0x0" ::: "memory");
#endif
}

// ---------------- denominator: den[b,v] = eps + sum_l m[b,v,l] ----------------
__global__ __launch_bounds__(256)
void dvs_denom_kernel(const float* __restrict__ centers,
                      const int* __restrict__ v_len,
                      const int* __restrict__ grid_thws,
                      float* __restrict__ denom) {
    const int b = blockIdx.x >> 3;
    const int v = blockIdx.x & 7;
    const int W = grid_thws[b * 3 + 2];
    const int H = grid_thws[b * 3 + 1];
    const int n = v_len[b];
    const float cx = centers[(b * NUM_VIEWS + v) * 2 + 0];
    const float cy = centers[(b * NUM_VIEWS + v) * 2 + 1];
    const float invW = 1.0f / (float)W;
    const float invH = 1.0f / (float)H;

    float s = 0.0f;
    for (int l = threadIdx.x; l < n; l += 256) {
        int i = l / W;
        int j = l - i * W;
        float x = ((float)j + 0.5f) * invW;
        float y = ((float)i + 0.5f) * invH;
        float dx = cx - x, dy = cy - y;
        s += __expf(-GAMMA * (dx * dx + dy * dy));
    }
    __shared__ float red[256];
    red[threadIdx.x] = s;
    __syncthreads();
    for (int off = 128; off > 0; off >>= 1) {
        if (threadIdx.x < (unsigned)off) red[threadIdx.x] += red[threadIdx.x + off];
        __syncthreads();
    }
    if (threadIdx.x == 0) denom[b * NUM_VIEWS + v] = red[0] + EPSV;
}

// ---------------- main: out[b,v,d] = (m @ v_pad)[v,d] / den[b,v] ----------------
__global__ __launch_bounds__(128)
void dvs_main_kernel(const float* __restrict__ v_pad,
                     const float* __restrict__ centers,
                     const int* __restrict__ v_len,
                     const int* __restrict__ grid_thws,
                     const float* __restrict__ denom,
                     float* __restrict__ out) {
    // double-buffered tiles
    __shared__ __align__(16) float sv[2 * CHUNK * SV_STRIDE];   // v_pad tile [l][col]
    __shared__ __align__(16) float smt[2 * 16 * SM_STRIDE];     // m tile [view][l], views 8..15 = 0
    __shared__ float cc[NUM_VIEWS * 2];

    const int b  = blockIdx.x >> 4;     // NCB == 16
    const int cb = blockIdx.x & 15;
    const int d0 = cb * COLS;

    const int tid  = threadIdx.x;
    const int wave = tid >> 5;
    const int lane = tid & 31;
    const int half = lane >> 4;         // 0: K{0,1}, 1: K{2,3}
    const int r    = lane & 15;         // M row (A) / N col (B)

    const int W = grid_thws[b * 3 + 2];
    const int n = v_len[b];
    const float invW = 1.0f / (float)W;
    const float invH = 1.0f / (float)grid_thws[b * 3 + 1];

    if (tid < NUM_VIEWS * 2) cc[tid] = centers[b * NUM_VIEWS * 2 + tid];
    __syncthreads();

    v8f acc = {0.f, 0.f, 0.f, 0.f, 0.f, 0.f, 0.f, 0.f};

    const float* vbase = v_pad + (size_t)b * LMAX * D_ + d0;
    const int colw = wave * 16 + r;
    const int nchunks = (n + CHUNK - 1) / CHUNK;

    // ---- staging helpers (inlined lambdas) ----
    auto issue_async = [&](int c, int buf) {
        const float* gsrc = vbase + (size_t)c * CHUNK * D_;
        float* lbase = &sv[buf * CHUNK * SV_STRIDE];
#pragma unroll
        for (int k = 0; k < ASYNC_PER_THREAD; ++k) {
            int t   = tid + k * 128;
            int row = t >> 4;           // COLS/4 == 16 segments per row
            int seg = t & 15;
            unsigned lds_off = (unsigned)(size_t)(&lbase[row * SV_STRIDE + seg * 4]);
            async_load_b128(lds_off, gsrc + (size_t)row * D_ + seg * 4);
        }
    };
    auto compute_m = [&](int c, int buf) {
        const int l0 = c * CHUNK;
        float* mb = &smt[buf * 16 * SM_STRIDE];
        for (int t = tid; t < 16 * CHUNK; t += 128) {
            int v    = t >> 6;          // 0..15
            int loff = t & 63;
            int l    = l0 + loff;
            float mval = 0.0f;
            if (v < NUM_VIEWS && l < n) {
                int i = l / W;
                int j = l - i * W;
                float x = ((float)j + 0.5f) * invW;
                float y = ((float)i + 0.5f) * invH;
                float dx = cc[v * 2 + 0] - x;
                float dy = cc[v * 2 + 1] - y;
                mval = __expf(-GAMMA * (dx * dx + dy * dy));
            }
            mb[v * SM_STRIDE + loff] = mval;
        }
    };

    // prologue: start chunk 0
    issue_async(0, 0);
    compute_m(0, 0);

    for (int c = 0; c < nchunks; ++c) {
        const int cur = c & 1;
        if (c + 1 < nchunks) {
            issue_async(c + 1, cur ^ 1);   // prefetch next tile (ASYNCcnt += 8)
            compute_m(c + 1, cur ^ 1);     // VALU overlaps the async flight
            wait_asynccnt_8();             // own wave's chunk-c transfers landed
        } else {
            wait_asynccnt_0();
        }
        __syncthreads();                   // all waves' data + m tile visible

        const float* svp = &sv[cur * CHUNK * SV_STRIDE];
        const float* smp = &smt[cur * 16 * SM_STRIDE] + r * SM_STRIDE + 2 * half;
#pragma unroll
        for (int s = 0; s < 16; ++s) {
            int lk = s * 4 + 2 * half;
            v2f a = *(const v2f*)(smp + s * 4);          // ds_load_b64, imm offsets
            v2f bb;
            bb.x = svp[(lk    ) * SV_STRIDE + colw];     // fused ds_load_2addr
            bb.y = svp[(lk + 1) * SV_STRIDE + colw];
            acc = __builtin_amdgcn_wmma_f32_16x16x4_f32(
                false, a, false, bb, (short)0, acc, false, false);
        }
        __syncthreads();                   // before buffers are overwritten
    }

    // C/D layout: VGPR idx vv, lanes 0-15 -> (M=vv, N=lane). Views 0..7 == M 0..7.
    if (lane < 16) {
        const int dcol = d0 + wave * 16 + lane;
        float* ob = out + (size_t)b * NUM_VIEWS * D_ + dcol;
        const float* dnp = denom + b * NUM_VIEWS;
#pragma unroll
        for (int vv = 0; vv < NUM_VIEWS; ++vv) {
            ob[(size_t)vv * D_] = acc[vv] / dnp[vv];
        }
    }
}

extern "C" void kernel_launch(void* const* d_in, const int* in_sizes, int n_in,
                              void* d_out, int out_size, void* d_ws, size_t ws_size,
                              hipStream_t stream) {
    (void)in_sizes; (void)n_in; (void)out_size; (void)ws_size;
    const float* v_pad   = (const float*)d_in[0];
    const float* centers = (const float*)d_in[1];
    const int*   v_len   = (const int*)d_in[2];
    const int*   g_thws  = (const int*)d_in[3];
    float* out   = (float*)d_out;
    float* denom = (float*)d_ws;   // B_*NUM_VIEWS floats of scratch

    dvs_denom_kernel<<<B_ * NUM_VIEWS, 256, 0, stream>>>(centers, v_len, g_thws, denom);
    dvs_main_kernel<<<B_ * NCB, 128, 0, stream>>>(v_pad, centers, v_len, g_thws, denom, out);
}